// GraphAttentionLayer_24945170055939
// MI455X (gfx1250) — compile-verified
//
#include <hip/hip_runtime.h>
#include <math.h>
#include <stdint.h>

#define N_NODES 4096
#define IN_F    128
#define HEADS   8
#define HIDDEN  8
#define OUT_F   (HEADS*HIDDEN)   // 64
#define SLOPE   0.2f
#define PPITCH  20               // P-tile row pitch: 16B-aligned float4 slots,
                                 // 20*k mod 64 distinct for k=0..15 -> no bank conflicts

typedef __attribute__((ext_vector_type(2))) float v2f;
typedef __attribute__((ext_vector_type(8))) float v8f;

// ---------------------------------------------------------------------------
// Kernel 1: g = h @ W   (4096x128 @ 128x64) using V_WMMA_F32_16X16X4_F32.
// One wave per 16x16 output tile. grid = (4096/16)*(64/16) = 1024 waves.
// ISA 7.12.2 f32 layouts:
//   A 16x4 : lane = M + 16*(K>=2), vgpr j -> K = 2*(lane>>4) + j
//   B 4x16 : lane = N + 16*(K-half), vgpr j -> K = 2*(lane>>4) + j
//   C/D    : vgpr r -> row r (lanes 0-15) / r+8 (lanes 16-31), col = lane&15
// ---------------------------------------------------------------------------
__global__ void gat_gemm_g(const float* __restrict__ h,
                           const float* __restrict__ W,
                           float* __restrict__ g) {
  const int tile = blockIdx.x;
  const int tm = tile >> 2;        // 256 row tiles
  const int tn = tile & 3;         // 4 col tiles
  const int lane = threadIdx.x & 31;
  const int mn   = lane & 15;      // M for A, N for B/C/D
  const int kh   = lane >> 4;      // which K-half this lane holds

  const int row = tm * 16 + mn;
  const int col = tn * 16 + mn;

  v8f acc = {};
  for (int kk = 0; kk < IN_F; kk += 4) {
    const int kb = kk + 2 * kh;
    v2f a, b;
    a.x = h[row * IN_F + kb + 0];
    a.y = h[row * IN_F + kb + 1];
    b.x = W[(kb + 0) * OUT_F + col];
    b.y = W[(kb + 1) * OUT_F + col];
    acc = __builtin_amdgcn_wmma_f32_16x16x4_f32(false, a, false, b,
                                                (short)0, acc, false, false);
  }
  const int rbase = tm * 16 + kh * 8;
#pragma unroll
  for (int r = 0; r < 8; ++r)
    g[(rbase + r) * OUT_F + col] = acc[r];
}

// ---------------------------------------------------------------------------
// Kernel 2: per-node attention scores.
//   slN[n*8+h] = <g[n,h,:], a_l> ;  srT[h*4096+n] = <g[n,h,:], a_r>
// srT is head-major so the attention kernel streams it contiguously.
// ---------------------------------------------------------------------------
__global__ void gat_scores(const float* __restrict__ g,
                           const float* __restrict__ a_l,
                           const float* __restrict__ a_r,
                           float* __restrict__ slN,
                           float* __restrict__ srT) {
  const int idx = blockIdx.x * blockDim.x + threadIdx.x;
  if (idx >= N_NODES * HEADS) return;
  const int n  = idx >> 3;
  const int hh = idx & 7;
  const float* gp = g + n * OUT_F + hh * HIDDEN;
  float sl = 0.f, sr = 0.f;
#pragma unroll
  for (int d = 0; d < HIDDEN; ++d) {
    sl += gp[d] * a_l[d];
    sr += gp[d] * a_r[d];
  }
  slN[n * HEADS + hh] = sl;
  srT[hh * N_NODES + n] = sr;
}

// ---------------------------------------------------------------------------
// Kernel 3: masked softmax + aggregation, fused.
// One block (256 threads = 8 wave32) per 16-row i-tile; wave w owns head w.
// Pass 1: row max of sr over neighbors (leaky-ReLU is monotone, so
//         e_max = LR(sl_i + max sr_j)); coalesced u32 adjacency scan.
// Pass 2: per 16-j chunk, each lane converts two u32 adjacency words (4 j's each)
//         into unnormalized probs and stores each as one ds_store_b128, then
//         4x V_WMMA_F32_16X16X4_F32 against a G-chunk whose column 8 is
//         all-ones -> D[:,8] accumulates the softmax denominator for free.
// ---------------------------------------------------------------------------
__device__ __forceinline__ float leaky(float x) {
  return (x >= 0.f) ? x : SLOPE * x;
}

__device__ __forceinline__ float bload(const float* __restrict__ g,
                                       int j, int w, int n) {
  if (n < HIDDEN) return g[j * OUT_F + w * HIDDEN + n];
  return (n == HIDDEN) ? 1.0f : 0.0f;   // ones column -> row sums
}

__global__ void gat_attn(const uint8_t* __restrict__ adj,
                         const float* __restrict__ g,
                         const float* __restrict__ slN,
                         const float* __restrict__ srT,
                         float* __restrict__ out) {
  __shared__ float sl_s[16 * HEADS];          // sl for the 16 tile rows
  __shared__ float em_s[16 * HEADS];          // row-wise max score (post-LR)
  __shared__ float pbuf[HEADS][16 * PPITCH];  // per-wave 16x16 P tile

  const int w    = threadIdx.x >> 5;   // head / wave id
  const int lane = threadIdx.x & 31;
  const int mn   = lane & 15;
  const int kh   = lane >> 4;
  const int i0   = blockIdx.x * 16;

  if (threadIdx.x < 16 * HEADS)
    sl_s[threadIdx.x] = slN[i0 * HEADS + threadIdx.x];
  __syncthreads();

  // ---- Pass 1: neighbor max of sr per (row, head) -------------------------
  const uint32_t* adjw = (const uint32_t*)adj;
  const float4*   sr4  = (const float4*)(srT + w * N_NODES);
  for (int i = 0; i < 16; ++i) {
    const uint32_t* rowp = adjw + (size_t)(i0 + i) * (N_NODES / 4);
    __builtin_prefetch(rowp + (N_NODES / 4), 0, 0);   // next adjacency row
    float m = -3.4e38f;                                // self-edge guarantees hit
    for (int t = lane; t < N_NODES / 4; t += 32) {
      const uint32_t word = rowp[t];
      if (word) {
        const float4 s = sr4[t];
        if (word & 0x000000FFu) m = fmaxf(m, s.x);
        if (word & 0x0000FF00u) m = fmaxf(m, s.y);
        if (word & 0x00FF0000u) m = fmaxf(m, s.z);
        if (word & 0xFF000000u) m = fmaxf(m, s.w);
      }
    }
#pragma unroll
    for (int off = 16; off; off >>= 1)
      m = fmaxf(m, __shfl_xor(m, off, 32));
    if (lane == 0)
      em_s[i * HEADS + w] = leaky(sl_s[i * HEADS + w] + m);
  }
  __syncthreads();

  // ---- Pass 2: WMMA accumulation of p @ [G | 1] ---------------------------
  float* pw = &pbuf[w][0];
  v8f acc = {};
  for (int jc = 0; jc < N_NODES; jc += 16) {
    // 64 adjacency words cover the 16x16 tile; 2 words per lane.
#pragma unroll
    for (int half = 0; half < 2; ++half) {
      const int wi  = lane + half * 32;   // 0..63
      const int row = wi >> 2;            // 0..15
      const int jq  = wi & 3;             // 4-j group within chunk
      const uint32_t word =
          *(const uint32_t*)(adj + (size_t)(i0 + row) * N_NODES + jc + jq * 4);
      const float4 s = *(const float4*)(srT + w * N_NODES + jc + jq * 4);
      const float slv = sl_s[row * HEADS + w];
      const float emv = em_s[row * HEADS + w];
      float4 p;
      p.x = (word & 0x000000FFu) ? __expf(leaky(slv + s.x) - emv) : 0.f;
      p.y = (word & 0x0000FF00u) ? __expf(leaky(slv + s.y) - emv) : 0.f;
      p.z = (word & 0x00FF0000u) ? __expf(leaky(slv + s.z) - emv) : 0.f;
      p.w = (word & 0xFF000000u) ? __expf(leaky(slv + s.w) - emv) : 0.f;
      *(float4*)&pw[row * PPITCH + jq * 4] = p;   // aligned ds_store_b128
    }
    // 4 x K=4 f32 WMMAs over this chunk.
#pragma unroll
    for (int k0 = 0; k0 < 16; k0 += 4) {
      const int ka = k0 + 2 * kh;   // this lane's K pair within the chunk
      v2f a, b;
      a.x = pw[mn * PPITCH + ka + 0];
      a.y = pw[mn * PPITCH + ka + 1];
      b.x = bload(g, jc + ka + 0, w, mn);
      b.y = bload(g, jc + ka + 1, w, mn);
      acc = __builtin_amdgcn_wmma_f32_16x16x4_f32(false, a, false, b,
                                                  (short)0, acc, false, false);
    }
  }

  // ---- Normalize by column 8 (the row-sum) and store ----------------------
#pragma unroll
  for (int r = 0; r < 8; ++r) {
    const float v = acc[r];
    const float S = __shfl(v, (lane & 16) | 8, 32);   // D[row][8]
    if (mn < HIDDEN) {
      const int row = i0 + r + kh * 8;
      out[row * OUT_F + w * HIDDEN + mn] = v / S;
    }
  }
}

// ---------------------------------------------------------------------------
extern "C" void kernel_launch(void* const* d_in, const int* in_sizes, int n_in,
                              void* d_out, int out_size, void* d_ws, size_t ws_size,
                              hipStream_t stream) {
  (void)in_sizes; (void)n_in; (void)out_size; (void)ws_size;
  const float*   h   = (const float*)d_in[0];
  const uint8_t* adj = (const uint8_t*)d_in[1];   // bool -> 1 byte
  const float*   W   = (const float*)d_in[2];
  const float*   a_l = (const float*)d_in[3];
  const float*   a_r = (const float*)d_in[4];
  float* out = (float*)d_out;

  float* g   = (float*)d_ws;                       // 4096*64
  float* srT = g + (size_t)N_NODES * OUT_F;        // 8*4096 (head-major)
  float* slN = srT + (size_t)HEADS * N_NODES;      // 4096*8

  gat_gemm_g<<<(N_NODES / 16) * (OUT_F / 16), 32, 0, stream>>>(h, W, g);
  gat_scores<<<(N_NODES * HEADS + 255) / 256, 256, 0, stream>>>(g, a_l, a_r, slN, srT);
  gat_attn<<<N_NODES / 16, 256, 0, stream>>>(adj, g, slN, srT, out);
}